// BindingConstraintsAB_137438954252
// MI455X (gfx1250) — compile-verified
//
#include <hip/hip_runtime.h>
#include <hip/hip_bf16.h>

// ---------------------------------------------------------------------------
// BindingConstraintsAB on MI455X (gfx1250, wave32).
//
// Algebraic restructuring:
//   x       = y @ Wp                       (only cols 0..8 ever used)
//   lam_x   has only cols 3..8 nonzero     -> lamx6 [N,6]
//   (y - a*lam_y) @ Wp = x - a * (lamx6 @ M6),  M6 = Wu[3:9,:] @ Wp  (6x18->6x9 used)
//   y_out   = y - (sum_i alpha_i * lamx6_i) @ Wu[3:9,:]  = y + (-Lacc) @ Wu6
//
// Big GEMMs (x=y@Wp and the final y update) use V_WMMA_F32_16X16X4_F32.
// Line-search scalar state machine runs in single-wave kernels; all
// reductions are deterministic (LDS tree + fixed-order serial sum).
// ---------------------------------------------------------------------------

#define NDIM 9
#define N_ITER 10
#define MAX_LS 11
#define CONVERGED 1e-4f

typedef float v2f __attribute__((ext_vector_type(2)));
typedef float v8f __attribute__((ext_vector_type(8)));

__device__ __forceinline__ v8f wmma_f32(v2f a, v2f b, v8f c) {
  // D(16x16) = A(16x4) * B(4x16) + C, fp32, wave32.
  return __builtin_amdgcn_wmma_f32_16x16x4_f32(
      /*neg_a=*/false, a, /*neg_b=*/false, b,
      /*c_mod=*/(short)0, c, /*reuse_a=*/false, /*reuse_b=*/false);
}

// Scalar state layout (floats): [0]=alpha [1]=ct [2]=cnorm [3]=step_scale
//                               [4]=found [5]=succ0 [6]=done [7]=ls_a

// -------------------------------------------------------------- init --------
__global__ void init_kernel(float* state, float* Lacc, int n) {
  int i = blockIdx.x * blockDim.x + threadIdx.x;
  if (i == 0) {
    state[0] = 1.0f;                       // alpha
    for (int k = 1; k < 16; ++k) state[k] = 0.0f;
  }
  size_t total = (size_t)n * 8;
  for (size_t t = i; t < total; t += (size_t)gridDim.x * blockDim.x)
    Lacc[t] = 0.0f;
}

// M6[j][c] = sum_k Wu[(3+j)*256 + k] * Wp[k*18 + c], j=0..5, c=0..15 (stride 16)
__global__ void m6_kernel(const float* __restrict__ Wp,
                          const float* __restrict__ Wu,
                          float* __restrict__ M6) {
  int t = threadIdx.x;
  if (t >= 96) return;
  int j = t >> 4, c = t & 15;
  float s = 0.0f;
  for (int k = 0; k < 256; ++k)
    s += Wu[(3 + j) * 256 + k] * Wp[k * 18 + c];
  M6[t] = s;
}

// ---------------------------------------------------- x9 = (y @ Wp)[:, :16] -
// One wave per 16-row tile; 8 waves per block; Wp (cols 0..15) staged in LDS.
__global__ void gemm_x_kernel(const float* __restrict__ y,
                              const float* __restrict__ Wp,
                              float* __restrict__ x9, int nTiles) {
  __shared__ float wp_lds[256 * 16];
  for (int t = threadIdx.x; t < 256 * 16; t += blockDim.x) {
    int k = t >> 4, c = t & 15;
    wp_lds[t] = Wp[k * 18 + c];
  }
  __syncthreads();

  int wave = threadIdx.x >> 5;
  int lane = threadIdx.x & 31;
  int tile = blockIdx.x * (blockDim.x >> 5) + wave;
  if (tile >= nTiles) return;                 // wave-uniform: EXEC all-ones inside

  int row = lane & 15;
  int klo = (lane >> 4) * 2;                  // lanes 0-15: K={0,1}; 16-31: K={2,3}
  const float* yrow = y + (size_t)(tile * 16 + row) * 256 + klo;

  v8f acc;
#pragma unroll
  for (int v = 0; v < 8; ++v) acc[v] = 0.0f;

  int col = lane & 15;
  for (int k0 = 0; k0 < 256; k0 += 4) {
    v2f a, b;
    a.x = yrow[k0];
    a.y = yrow[k0 + 1];
    b.x = wp_lds[(k0 + klo) * 16 + col];      // B[k0+klo  ][col]
    b.y = wp_lds[(k0 + klo + 1) * 16 + col];  // B[k0+klo+1][col]
    acc = wmma_f32(a, b, acc);
  }

  // C/D layout: lanes 0-15 -> M=v, lanes 16-31 -> M=v+8; N = lane&15
  int mhi = (lane >> 4) * 8;
  float* out = x9 + (size_t)(tile * 16 + mhi) * 16 + col;
#pragma unroll
  for (int v = 0; v < 8; ++v) out[v * 16] = acc[v];
}

// ------------------------------------------- constraints + lamx6 + z9 + cnorm
__device__ __forceinline__ void constraint_pair(const float* r, float dab,
                                                float dan, float* cb, float* cn,
                                                float* db, float* dn) {
  float r0x = r[0], r0y = r[1], r0z = r[2];
  db[0] = r[3] - r0x; db[1] = r[4] - r0y; db[2] = r[5] - r0z;
  dn[0] = r[6] - r0x; dn[1] = r[7] - r0y; dn[2] = r[8] - r0z;
  float sb = db[0]*db[0] + db[1]*db[1] + db[2]*db[2];
  float sn = dn[0]*dn[0] + dn[1]*dn[1] + dn[2]*dn[2];
  *cb = sqrtf(dab * dab / sb) - 1.0f;
  *cn = sqrtf(dan * dan / sn) - 1.0f;
}

__device__ __forceinline__ void block_reduce_store(float local, float* partials,
                                                   float* red) {
  red[threadIdx.x] = local;
  __syncthreads();
  for (int s = blockDim.x >> 1; s > 0; s >>= 1) {
    if ((int)threadIdx.x < s) red[threadIdx.x] += red[threadIdx.x + s];
    __syncthreads();
  }
  if (threadIdx.x == 0) partials[blockIdx.x] = red[0];
}

__global__ void constraints_kernel(const float* __restrict__ x9,
                                   const float* __restrict__ d_ab,
                                   const float* __restrict__ d_an,
                                   const float* __restrict__ M6,
                                   float* __restrict__ lamx6,
                                   float* __restrict__ z9,
                                   float* __restrict__ partials, int n) {
  __shared__ float red[256];
  __shared__ float m6s[96];
  for (int t = threadIdx.x; t < 96; t += blockDim.x) m6s[t] = M6[t];
  __syncthreads();

  int i = blockIdx.x * blockDim.x + threadIdx.x;
  float local = 0.0f;
  if (i < n) {
    float r[9];
    const float* xr = x9 + (size_t)i * 16;
#pragma unroll
    for (int c = 0; c < 9; ++c) r[c] = xr[c];
    float cb, cn, db[3], dn[3];
    constraint_pair(r, d_ab[i], d_an[i], &cb, &cn, db, dn);
    float l[6] = {-db[0]*cb, -db[1]*cb, -db[2]*cb,
                  -dn[0]*cn, -dn[1]*cn, -dn[2]*cn};
    float* lr = lamx6 + (size_t)i * 8;
#pragma unroll
    for (int j = 0; j < 6; ++j) lr[j] = l[j];
    float* zr = z9 + (size_t)i * 16;
#pragma unroll
    for (int c = 0; c < 9; ++c) {
      float s = 0.0f;
#pragma unroll
      for (int j = 0; j < 6; ++j) s += l[j] * m6s[j * 16 + c];
      zr[c] = s;
    }
    local = cb * cb + cn * cn;
  }
  block_reduce_store(local, partials, red);
}

// cnorm = sum(partials); reset line-search state
__global__ void reduce_cnorm_kernel(float* state, const float* partials,
                                    int nb) {
  if (threadIdx.x == 0) {
    float s = 0.0f;
    for (int i = 0; i < nb; ++i) s += partials[i];
    state[2] = s;                  // cnorm
    state[7] = state[0];           // ls_a = alpha
    state[4] = 0.0f;               // found
    state[1] = __builtin_inff();   // ct
    state[5] = 0.0f;               // succ0
  }
}

// cnorm_of(a) partials with xt = x9 - a * z9
__global__ void ls_eval_kernel(const float* __restrict__ x9,
                               const float* __restrict__ z9,
                               const float* __restrict__ d_ab,
                               const float* __restrict__ d_an,
                               const float* __restrict__ state,
                               float* __restrict__ partials, int n) {
  __shared__ float red[256];
  float a = state[7];
  int i = blockIdx.x * blockDim.x + threadIdx.x;
  float local = 0.0f;
  if (i < n) {
    float r[9];
    const float* xr = x9 + (size_t)i * 16;
    const float* zr = z9 + (size_t)i * 16;
#pragma unroll
    for (int c = 0; c < 9; ++c) r[c] = xr[c] - a * zr[c];
    float cb, cn, db[3], dn[3];
    constraint_pair(r, d_ab[i], d_an[i], &cb, &cn, db, dn);
    local = cb * cb + cn * cn;
  }
  block_reduce_store(local, partials, red);
}

// One line-search scan step (scalar state machine)
__global__ void ls_finish_kernel(float* state, const float* partials, int nb,
                                 int step) {
  if (threadIdx.x == 0) {
    float ct_new = 0.0f;
    for (int i = 0; i < nb; ++i) ct_new += partials[i];
    bool found = state[4] > 0.5f;
    bool succ = ct_new < state[2];
    if (step == 0) state[5] = succ ? 1.0f : 0.0f;     // succs[0]
    if (!(found || succ)) state[7] *= 0.5f;           // a_new
    if (!found) state[1] = ct_new;                    // ct_keep
    if (succ) state[4] = 1.0f;                        // found |= succ
  }
}

// End of outer iteration: alpha growth, done flag, masked step scale
__global__ void outer_finish_kernel(float* state) {
  if (threadIdx.x == 0) {
    float a_f = state[7], ctry = state[1];
    float alpha_new = (state[5] > 0.5f && ctry > CONVERGED) ? a_f * 1.5f : a_f;
    bool done = state[6] > 0.5f;
    state[3] = done ? 0.0f : alpha_new;   // step_scale (masked update)
    if (!done) state[0] = alpha_new;      // carried alpha
    if (ctry < CONVERGED) state[6] = 1.0f;
  }
}

// x9 -= ss*z9 ; Lacc += ss*lamx6
__global__ void update_kernel(const float* __restrict__ state,
                              float* __restrict__ x9,
                              const float* __restrict__ z9,
                              const float* __restrict__ lamx6,
                              float* __restrict__ Lacc, int n) {
  float ss = state[3];
  int i = blockIdx.x * blockDim.x + threadIdx.x;
  if (i >= n) return;
  float* xr = x9 + (size_t)i * 16;
  const float* zr = z9 + (size_t)i * 16;
#pragma unroll
  for (int c = 0; c < 9; ++c) xr[c] -= ss * zr[c];
  float* La = Lacc + (size_t)i * 8;
  const float* lr = lamx6 + (size_t)i * 8;
#pragma unroll
  for (int j = 0; j < 6; ++j) La[j] += ss * lr[j];
}

// --------------------------------------- y_out = y + (-Lacc) @ Wu[3:9,:] ----
// One wave per 16-row tile, looping 16 column tiles; C seeded from y.
__global__ void gemm_out_kernel(const float* __restrict__ y,
                                const float* __restrict__ Wu,
                                const float* __restrict__ Lacc,
                                float* __restrict__ yout, int nTiles) {
  int wave = threadIdx.x >> 5;
  int lane = threadIdx.x & 31;
  int tile = blockIdx.x * (blockDim.x >> 5) + wave;
  if (tile >= nTiles) return;

  int row = lane & 15;
  int klo = (lane >> 4) * 2;
  int col = lane & 15;
  int mhi = (lane >> 4) * 8;
  bool lo = lane < 16;

  const float* lrow = Lacc + (size_t)(tile * 16 + row) * 8;
  v2f a0, a1;
  a0.x = -lrow[klo];                        // K-step 0: k = klo, klo+1
  a0.y = -lrow[klo + 1];
  a1.x = lo ? -lrow[4] : 0.0f;              // K-step 1: k = 4,5 (6,7 padded 0)
  a1.y = lo ? -lrow[5] : 0.0f;

  for (int ct = 0; ct < 16; ++ct) {
    int cb = ct * 16 + col;
    v2f b0, b1;
    b0.x = Wu[(3 + klo) * 256 + cb];        // Wu6 rows klo, klo+1
    b0.y = Wu[(3 + klo + 1) * 256 + cb];
    b1.x = lo ? Wu[7 * 256 + cb] : 0.0f;    // Wu6 rows 4,5 ; rows 6,7 = 0 pad
    b1.y = lo ? Wu[8 * 256 + cb] : 0.0f;

    v8f c;
    const float* yc = y + (size_t)(tile * 16 + mhi) * 256 + cb;
#pragma unroll
    for (int v = 0; v < 8; ++v) c[v] = yc[v * 256];

    c = wmma_f32(a0, b0, c);
    c = wmma_f32(a1, b1, c);

    float* oc = yout + (size_t)(tile * 16 + mhi) * 256 + cb;
#pragma unroll
    for (int v = 0; v < 8; ++v) oc[v * 256] = c[v];
  }
}

// ---------------------------------------------------------------------------
extern "C" void kernel_launch(void* const* d_in, const int* in_sizes, int n_in,
                              void* d_out, int out_size, void* d_ws,
                              size_t ws_size, hipStream_t stream) {
  const float* y    = (const float*)d_in[0];
  const float* d_ab = (const float*)d_in[2];
  const float* d_an = (const float*)d_in[3];
  const float* Wp   = (const float*)d_in[5];
  const float* Wu   = (const float*)d_in[6];
  float* yout = (float*)d_out;

  int N = in_sizes[2];                    // 100000 nodes
  int nTiles = (N + 15) / 16;             // 6250 (N divisible by 16)
  int nodeBlocks = (N + 255) / 256;       // 391
  int gemmBlocks = (nTiles + 7) / 8;      // 8 waves per block

  // Workspace carve-up (floats): ~19.3 MB total
  float* ws      = (float*)d_ws;
  float* state   = ws;                    // 16
  float* partials= ws + 16;               // 1024 slots
  float* M6      = ws + 1040;             // 96 (pad to 128)
  float* x9      = ws + 1168;             // N*16
  float* z9      = x9 + (size_t)N * 16;   // N*16
  float* lamx6   = z9 + (size_t)N * 16;   // N*8
  float* Lacc    = lamx6 + (size_t)N * 8; // N*8

  init_kernel<<<nodeBlocks, 256, 0, stream>>>(state, Lacc, N);
  m6_kernel<<<1, 128, 0, stream>>>(Wp, Wu, M6);
  gemm_x_kernel<<<gemmBlocks, 256, 0, stream>>>(y, Wp, x9, nTiles);

  for (int it = 0; it < N_ITER; ++it) {
    constraints_kernel<<<nodeBlocks, 256, 0, stream>>>(x9, d_ab, d_an, M6,
                                                       lamx6, z9, partials, N);
    reduce_cnorm_kernel<<<1, 32, 0, stream>>>(state, partials, nodeBlocks);
    for (int s = 0; s < MAX_LS; ++s) {
      ls_eval_kernel<<<nodeBlocks, 256, 0, stream>>>(x9, z9, d_ab, d_an, state,
                                                     partials, N);
      ls_finish_kernel<<<1, 32, 0, stream>>>(state, partials, nodeBlocks, s);
    }
    outer_finish_kernel<<<1, 32, 0, stream>>>(state);
    update_kernel<<<nodeBlocks, 256, 0, stream>>>(state, x9, z9, lamx6, Lacc, N);
  }

  gemm_out_kernel<<<gemmBlocks, 256, 0, stream>>>(y, Wu, Lacc, yout, nTiles);
}